// Graph_Layer_gatHeadv2_36507222016270
// MI455X (gfx1250) — compile-verified
//
#include <hip/hip_runtime.h>
#include <math.h>

#define NN 50000
#define NE 600000
#define DD 128
#define NG 8

typedef __attribute__((ext_vector_type(2))) float v2f;
typedef __attribute__((ext_vector_type(8))) float v8f;

// ---------------------------------------------------------------------------
// out[row][col] = sum_k X[row][k] * W[col][k] + b[col]      (X @ W^T + b)
// One wave32 per 16-row tile. WMMA f32 16x16x4, K-loop of 32 steps,
// 8 accumulators cover the 128 output columns.
// A layout (16x4 f32): lanes 0-15 hold K=0..1 (v0,v1), lanes 16-31 K=2..3.
// B layout (4x16 f32): mirrored; B[k][n] = W[n][k] -> per-lane float2 load.
// ---------------------------------------------------------------------------
__global__ __launch_bounds__(32) void gemm_xWT(const float* __restrict__ X,
                                               const float* __restrict__ W,
                                               const float* __restrict__ b,
                                               float* __restrict__ out) {
  const int lane = threadIdx.x & 31;
  const int half = lane >> 4;
  const int lm   = lane & 15;
  const int row0 = blockIdx.x * 16;

  v8f acc[8];
#pragma unroll
  for (int nt = 0; nt < 8; ++nt) { v8f z = {}; acc[nt] = z; }

  const float* arow = X + (size_t)(row0 + lm) * DD + 2 * half;
  for (int kk = 0; kk < DD / 4; ++kk) {
    v2f a = *(const v2f*)(arow + kk * 4);
#pragma unroll
    for (int nt = 0; nt < 8; ++nt) {
      v2f bb = *(const v2f*)(W + (size_t)(nt * 16 + lm) * DD + kk * 4 + 2 * half);
      acc[nt] = __builtin_amdgcn_wmma_f32_16x16x4_f32(
          false, a, false, bb, (short)0, acc[nt], false, false);
    }
  }
  // C/D layout: VGPR r, lanes 0-15 -> M=r, lanes 16-31 -> M=r+8, N=lane%16
#pragma unroll
  for (int nt = 0; nt < 8; ++nt) {
    const int col = nt * 16 + lm;
    const float bias = b[col];
#pragma unroll
    for (int r = 0; r < 8; ++r)
      out[(size_t)(row0 + r + 8 * half) * DD + col] = acc[nt][r] + bias;
  }
}

// ---------------------------------------------------------------------------
// Per 16-edge tile: e = edge_attr @ We^T via WMMA (kept in registers),
// m = leaky_relu(xl[src] + xr[dst] + e), alpha = m . att,
// then segment-max via float atomicMax.
// ---------------------------------------------------------------------------
__global__ __launch_bounds__(32) void edge_alpha(const float* __restrict__ EA,
                                                 const float* __restrict__ We,
                                                 const int* __restrict__ src,
                                                 const int* __restrict__ dst,
                                                 const float* __restrict__ xl,
                                                 const float* __restrict__ xr,
                                                 const float* __restrict__ att,
                                                 float* __restrict__ alpha,
                                                 float* __restrict__ amax) {
  const int lane = threadIdx.x & 31;
  const int half = lane >> 4;
  const int lm   = lane & 15;
  const int row0 = blockIdx.x * 16;

  v8f acc[8];
#pragma unroll
  for (int nt = 0; nt < 8; ++nt) { v8f z = {}; acc[nt] = z; }

  const float* arow = EA + (size_t)(row0 + lm) * DD + 2 * half;
  for (int kk = 0; kk < DD / 4; ++kk) {
    v2f a = *(const v2f*)(arow + kk * 4);
#pragma unroll
    for (int nt = 0; nt < 8; ++nt) {
      v2f bb = *(const v2f*)(We + (size_t)(nt * 16 + lm) * DD + kk * 4 + 2 * half);
      acc[nt] = __builtin_amdgcn_wmma_f32_16x16x4_f32(
          false, a, false, bb, (short)0, acc[nt], false, false);
    }
  }

  int si[8], di[8];
#pragma unroll
  for (int r = 0; r < 8; ++r) {
    si[r] = src[row0 + r + 8 * half];
    di[r] = dst[row0 + r + 8 * half];
  }

  float s[8];
#pragma unroll
  for (int r = 0; r < 8; ++r) s[r] = 0.0f;

#pragma unroll
  for (int nt = 0; nt < 8; ++nt) {
    const int col = nt * 16 + lm;
    const float av = att[col];
#pragma unroll
    for (int r = 0; r < 8; ++r) {
      float m = xl[(size_t)si[r] * DD + col] + xr[(size_t)di[r] * DD + col] + acc[nt][r];
      m = (m > 0.0f) ? m : 0.2f * m;        // leaky_relu, slope 0.2
      s[r] += m * av;
    }
  }
  // reduce each row-partial across the 16 lanes of each half-wave
#pragma unroll
  for (int mask = 1; mask <= 8; mask <<= 1) {
#pragma unroll
    for (int r = 0; r < 8; ++r) s[r] += __shfl_xor(s[r], mask, 32);
  }
  if (lm == 0) {                            // lanes 0 and 16
#pragma unroll
    for (int r = 0; r < 8; ++r) {
      alpha[row0 + r + 8 * half] = s[r];
      atomicMax(&amax[di[r]], s[r]);        // global_atomic_max_num_f32
    }
  }
}

__global__ void edge_exp(float* __restrict__ alpha, const int* __restrict__ dst,
                         const float* __restrict__ amax, float* __restrict__ denom) {
  int e = blockIdx.x * blockDim.x + threadIdx.x;
  if (e >= NE) return;
  int d = dst[e];
  float ex = __expf(alpha[e] - amax[d]);
  alpha[e] = ex;                            // in-place: raw alpha -> exp(alpha - max)
  atomicAdd(&denom[d], ex);
}

// one wave per edge: yacc[dst] += xl[src] * (ea / denom[dst])
__global__ void edge_aggr(const int* __restrict__ src, const int* __restrict__ dst,
                          const float* __restrict__ xl, const float* __restrict__ ea,
                          const float* __restrict__ denom, float* __restrict__ yacc) {
  int wave = (int)((blockIdx.x * (size_t)blockDim.x + threadIdx.x) >> 5);
  int lane = threadIdx.x & 31;
  if (wave >= NE) return;
  int s = src[wave], d = dst[wave];
  float w = ea[wave] / (denom[d] + 1e-16f);
  const float4 v = *(const float4*)(xl + (size_t)s * DD + lane * 4);
  float* yd = yacc + (size_t)d * DD + lane * 4;
  atomicAdd(&yd[0], v.x * w);
  atomicAdd(&yd[1], v.y * w);
  atomicAdd(&yd[2], v.z * w);
  atomicAdd(&yd[3], v.w * w);
}

// one wave per node: per-graph sum / sumsq / count
__global__ void node_stats(const float* __restrict__ yacc, const int* __restrict__ batch,
                           float* __restrict__ gsum, float* __restrict__ gsumsq,
                           float* __restrict__ gcnt) {
  int wave = (int)((blockIdx.x * (size_t)blockDim.x + threadIdx.x) >> 5);
  int lane = threadIdx.x & 31;
  if (wave >= NN) return;
  const float4 v = *(const float4*)(yacc + (size_t)wave * DD + lane * 4);
  float s = v.x + v.y + v.z + v.w;
  float q = v.x * v.x + v.y * v.y + v.z * v.z + v.w * v.w;
#pragma unroll
  for (int mask = 1; mask < 32; mask <<= 1) {
    s += __shfl_xor(s, mask, 32);
    q += __shfl_xor(q, mask, 32);
  }
  if (lane == 0) {
    int g = batch[wave];
    atomicAdd(&gsum[g], s);
    atomicAdd(&gsumsq[g], q);
    atomicAdd(&gcnt[g], 1.0f);
  }
}

__global__ void graph_stats(const float* __restrict__ gsum, const float* __restrict__ gsumsq,
                            const float* __restrict__ gcnt, float* __restrict__ gmean,
                            float* __restrict__ grstd) {
  int g = threadIdx.x;
  if (g >= NG) return;
  float norm = fmaxf(gcnt[g], 1.0f) * (float)DD;
  float mean = gsum[g] / norm;
  float var  = fmaxf(gsumsq[g] / norm - mean * mean, 0.0f);
  gmean[g] = mean;
  grstd[g] = rsqrtf(var + 1e-5f);
}

__global__ void finalize(const float* __restrict__ yacc, const int* __restrict__ batch,
                         const float* __restrict__ gmean, const float* __restrict__ grstd,
                         const float* __restrict__ lw, const float* __restrict__ lb,
                         float* __restrict__ out) {
  size_t i = blockIdx.x * (size_t)blockDim.x + threadIdx.x;
  if (i >= (size_t)NN * DD) return;
  int n = (int)(i >> 7), c = (int)(i & 127);
  int g = batch[n];
  out[i] = (yacc[i] - gmean[g]) * grstd[g] * lw[c] + lb[c];
}

__global__ void init_y(const float* __restrict__ x, const float* __restrict__ bias,
                       float* __restrict__ yacc) {
  size_t i = blockIdx.x * (size_t)blockDim.x + threadIdx.x;
  if (i < (size_t)NN * DD) yacc[i] = x[i] + bias[i & 127];   // residual base: x + out_bias
}

__global__ void init_node(float* __restrict__ amax, float* __restrict__ denom,
                          float* __restrict__ g5) {
  int i = blockIdx.x * blockDim.x + threadIdx.x;
  if (i < NN) { amax[i] = -__builtin_inff(); denom[i] = 0.0f; }
  if (i < 5 * NG) g5[i] = 0.0f;
}

extern "C" void kernel_launch(void* const* d_in, const int* in_sizes, int n_in,
                              void* d_out, int out_size, void* d_ws, size_t ws_size,
                              hipStream_t stream) {
  const float* x     = (const float*)d_in[0];
  const int*   ei    = (const int*)d_in[1];      // [2, NE] int32 (JAX x64 disabled)
  const float* eattr = (const float*)d_in[2];
  // d_in[3] = memory (unused by reference output)
  const int*   batch = (const int*)d_in[4];
  const float* Wl = (const float*)d_in[5];
  const float* bl = (const float*)d_in[6];
  const float* Wr = (const float*)d_in[7];
  const float* br = (const float*)d_in[8];
  const float* We = (const float*)d_in[9];
  const float* att  = (const float*)d_in[10];
  const float* bias = (const float*)d_in[11];
  const float* lnw  = (const float*)d_in[12];
  const float* lnb  = (const float*)d_in[13];

  const int* src = ei;            // edge_index[0]
  const int* dst = ei + NE;       // edge_index[1]

  // workspace layout (floats), ~80 MB total
  float* xl    = (float*)d_ws;
  float* xr    = xl + (size_t)NN * DD;
  float* yacc  = xr + (size_t)NN * DD;
  float* alpha = yacc + (size_t)NN * DD;  // NE, reused as ea after edge_exp
  float* amax  = alpha + NE;              // NN
  float* denom = amax + NN;               // NN
  float* gsum  = denom + NN;              // NG
  float* gsq   = gsum + NG;
  float* gcnt  = gsq + NG;
  float* gmean = gcnt + NG;
  float* grstd = gmean + NG;

  float* out_x = (float*)d_out;                    // [NN, DD]
  float* out_e = out_x + (size_t)NN * DD;          // [NE, DD] passthrough

  // init
  init_y<<<(NN * DD + 255) / 256, 256, 0, stream>>>(x, bias, yacc);
  init_node<<<(NN + 255) / 256, 256, 0, stream>>>(amax, denom, gsum);

  // node transforms via WMMA f32 16x16x4
  gemm_xWT<<<NN / 16, 32, 0, stream>>>(x, Wl, bl, xl);
  gemm_xWT<<<NN / 16, 32, 0, stream>>>(x, Wr, br, xr);

  // per-edge attention logits + segment max (edge GEMM fused, never hits HBM)
  edge_alpha<<<NE / 16, 32, 0, stream>>>(eattr, We, src, dst, xl, xr, att, alpha, amax);

  // softmax numerator + denominator
  edge_exp<<<(NE + 255) / 256, 256, 0, stream>>>(alpha, dst, amax, denom);

  // weighted aggregation into residual accumulator
  edge_aggr<<<NE / 8, 256, 0, stream>>>(src, dst, xl, alpha, denom, yacc);

  // graph-mode LayerNorm
  node_stats<<<NN / 8, 256, 0, stream>>>(yacc, batch, gsum, gsq, gcnt);
  graph_stats<<<1, NG, 0, stream>>>(gsum, gsq, gcnt, gmean, grstd);
  finalize<<<(NN * DD + 255) / 256, 256, 0, stream>>>(yacc, batch, gmean, grstd, lnw, lnb, out_x);

  // second output: edge_attr passthrough
  hipMemcpyAsync(out_e, eattr, (size_t)NE * DD * sizeof(float),
                 hipMemcpyDeviceToDevice, stream);
}